// Encoder_70935679861344
// MI455X (gfx1250) — compile-verified
//
#include <hip/hip_runtime.h>
#include <hip/hip_bf16.h>
#include <stdint.h>

// ---------------------------------------------------------------------------
// GCN encoder for MI455X (gfx1250, wave32).
// Dense GEMMs use V_WMMA_F32_16X16X4_F32 (exact fp32 matrix pipe).
// A-tile staging uses GLOBAL_LOAD_ASYNC_TO_LDS_B128 (ASYNCcnt path).
// Edge scatter uses f32 global atomics (memory-bound part of the op).
// ---------------------------------------------------------------------------

typedef __attribute__((ext_vector_type(2))) float v2f;
typedef __attribute__((ext_vector_type(8))) float v8f;

#define CH 128          // IN_CH == HID == 128
#define OUTCH 64

// ---------------- degree / normalization ----------------

__global__ __launch_bounds__(256) void deg_init_kernel(float* deg, int n) {
    int i = blockIdx.x * 256 + threadIdx.x;
    if (i < n) deg[i] = 1.0f;                 // self-loop
}

__global__ __launch_bounds__(256) void deg_edges_kernel(const int* __restrict__ dst,
                                                        float* deg, int e) {
    int i = blockIdx.x * 256 + threadIdx.x;
    if (i < e) atomicAdd(&deg[dst[i]], 1.0f);
}

__global__ __launch_bounds__(256) void dinv_kernel(float* d, int n) {
    int i = blockIdx.x * 256 + threadIdx.x;
    if (i < n) d[i] = rsqrtf(fmaxf(d[i], 1.0f));
}

// ---------------- WMMA node GEMM: H = X @ W  (K = F = 128) ----------------
// Block = 256 threads = 8 waves. Block computes 16 rows x 128 cols.
// Wave w owns column tile [16w, 16w+16). A tile (16x128) staged in LDS via
// async global->LDS B128 copies (ASYNCcnt), then consumed with ds loads.

__global__ __launch_bounds__(256) void gemm_node_wmma(const float* __restrict__ X,
                                                      const float* __restrict__ W,
                                                      float* __restrict__ H, int N) {
    __shared__ float ldsA[16 * CH];           // 8 KB
    const int rowBase = blockIdx.x * 16;
    const int tid = threadIdx.x;

    // async stage of 16 rows of X: 8192 B = 512 x B128, 2 per thread
#pragma unroll
    for (int i = tid * 4; i < 16 * CH; i += 256 * 4) {
        int r = i >> 7;                       // i / 128
        int k = i & (CH - 1);
        int row = rowBase + r; if (row >= N) row = N - 1;
        uint64_t gaddr = (uint64_t)(const void*)&X[(long)row * CH + k];
        uint32_t laddr = (uint32_t)(uintptr_t)&ldsA[i];
        asm volatile("global_load_async_to_lds_b128 %0, %1, off"
                     :: "v"(laddr), "v"(gaddr) : "memory");
    }
    asm volatile("s_wait_asynccnt 0x0" ::: "memory");
    __syncthreads();

    const int wave = tid >> 5;
    const int lane = tid & 31;
    const int colBase = wave * 16;            // F == 128 -> 8 waves cover all cols
    const int r  = lane & 15;                 // A row within tile
    const int kk = (lane >> 4) * 2;           // lanes 0-15: K 0,1 ; lanes 16-31: K 2,3
    const int c  = colBase + (lane & 15);     // B/D column

    v8f acc = {};
    for (int k = 0; k < CH; k += 4) {
        v2f a, b;
        a.x = ldsA[r * CH + k + kk];
        a.y = ldsA[r * CH + k + kk + 1];
        b.x = W[(long)(k + kk) * CH + c];
        b.y = W[(long)(k + kk + 1) * CH + c];
        acc = __builtin_amdgcn_wmma_f32_16x16x4_f32(false, a, false, b,
                                                    (short)0, acc, false, false);
    }
    // C/D layout: VGPR i -> M = i (lanes 0-15) / i+8 (lanes 16-31)
    const int row0 = rowBase + (lane >> 4) * 8;
#pragma unroll
    for (int i = 0; i < 8; ++i) {
        int row = row0 + i;
        if (row < N) H[(long)row * CH + c] = acc[i];
    }
}

// ---------------- aggregation (self loop + edges) ----------------

__global__ __launch_bounds__(256) void agg_init_kernel(const float* __restrict__ h,
                                                       const float* __restrict__ dinv,
                                                       float* __restrict__ agg, int n) {
    int t = blockIdx.x * 256 + threadIdx.x;   // n*32 threads, float4 each
    if (t >= n * 32) return;
    int node = t >> 5;
    int c4 = (t & 31) * 4;
    float s = dinv[node]; s = s * s;          // self-loop norm = dinv^2
    float4 v = *(const float4*)&h[(long)node * CH + c4];
    v.x *= s; v.y *= s; v.z *= s; v.w *= s;
    *(float4*)&agg[(long)node * CH + c4] = v;
}

__global__ __launch_bounds__(256) void agg_edges_kernel(const int* __restrict__ src,
                                                        const int* __restrict__ dst,
                                                        const float* __restrict__ h,
                                                        const float* __restrict__ dinv,
                                                        float* __restrict__ agg, int e) {
    int widx = (blockIdx.x * 256 + threadIdx.x) >> 5;  // one wave per edge
    int lane = threadIdx.x & 31;
    if (widx >= e) return;
    int s = src[widx], d = dst[widx];
    float nrm = dinv[s] * dinv[d];
    int c4 = lane * 4;
    float4 v = *(const float4*)&h[(long)s * CH + c4];
    float* o = &agg[(long)d * CH + c4];
    atomicAdd(o + 0, v.x * nrm);
    atomicAdd(o + 1, v.y * nrm);
    atomicAdd(o + 2, v.z * nrm);
    atomicAdd(o + 3, v.w * nrm);
}

__global__ __launch_bounds__(256) void bias_relu_kernel(float* a, const float* __restrict__ b,
                                                        int n) {
    int t = blockIdx.x * 256 + threadIdx.x;
    if (t >= n * CH) return;
    float v = a[t] + b[t & (CH - 1)];
    a[t] = v > 0.0f ? v : 0.0f;
}

// ---------------- pooling ----------------

__global__ __launch_bounds__(256) void pool_zero_kernel(float* sums, float* cnts, int g) {
    int t = blockIdx.x * 256 + threadIdx.x;
    if (t < g * CH) sums[t] = 0.0f;
    if (t < g) cnts[t] = 0.0f;
}

__global__ __launch_bounds__(256) void pool_accum_kernel(const float* __restrict__ h,
                                                         const int* __restrict__ batch,
                                                         float* sums, float* cnts, int n) {
    int widx = (blockIdx.x * 256 + threadIdx.x) >> 5;  // one wave per node
    int lane = threadIdx.x & 31;
    if (widx >= n) return;
    int g = batch[widx];
    int c4 = lane * 4;
    float4 v = *(const float4*)&h[(long)widx * CH + c4];
    float* o = &sums[(long)g * CH + c4];
    atomicAdd(o + 0, v.x);
    atomicAdd(o + 1, v.y);
    atomicAdd(o + 2, v.z);
    atomicAdd(o + 3, v.w);
    if (lane == 0) atomicAdd(&cnts[g], 1.0f);
}

__global__ __launch_bounds__(256) void pool_div_kernel(float* sums, const float* __restrict__ cnts,
                                                       int g) {
    int t = blockIdx.x * 256 + threadIdx.x;
    if (t >= g * CH) return;
    sums[t] /= fmaxf(cnts[t >> 7], 1.0f);
}

// ---------------- WMMA head GEMM: mu / logvar = pooled @ W + b ----------------
// G x 128 @ 128 x 64. Block = 8 waves: wave>>2 selects matrix, wave&3 the col tile.

__global__ __launch_bounds__(256) void head_gemm_wmma(const float* __restrict__ P,
                                                      const float* __restrict__ Wmu,
                                                      const float* __restrict__ bmu,
                                                      const float* __restrict__ Wlv,
                                                      const float* __restrict__ blv,
                                                      float* __restrict__ out, int G) {
    const int rowBase = blockIdx.x * 16;
    const int wave = threadIdx.x >> 5;
    const int lane = threadIdx.x & 31;
    const int mat = wave >> 2;                // 0 = mu, 1 = logvar
    const int colBase = (wave & 3) * 16;      // OUTCH = 64 -> 4 tiles
    const float* W = mat ? Wlv : Wmu;
    const float* b = mat ? blv : bmu;
    float* O = out + (long)mat * G * OUTCH;

    const int r  = lane & 15;
    const int kk = (lane >> 4) * 2;
    const int c  = colBase + (lane & 15);

    v8f acc = {};
    for (int k = 0; k < CH; k += 4) {
        v2f a, bb;
        int row = rowBase + r; if (row >= G) row = G - 1;
        a.x = P[(long)row * CH + k + kk];
        a.y = P[(long)row * CH + k + kk + 1];
        bb.x = W[(long)(k + kk) * OUTCH + c];
        bb.y = W[(long)(k + kk + 1) * OUTCH + c];
        acc = __builtin_amdgcn_wmma_f32_16x16x4_f32(false, a, false, bb,
                                                    (short)0, acc, false, false);
    }
    float bias = b[c];
    const int row0 = rowBase + (lane >> 4) * 8;
#pragma unroll
    for (int i = 0; i < 8; ++i) {
        int row = row0 + i;
        if (row < G) O[(long)row * OUTCH + c] = acc[i] + bias;
    }
}

// ---------------------------------------------------------------------------

extern "C" void kernel_launch(void* const* d_in, const int* in_sizes, int n_in,
                              void* d_out, int out_size, void* d_ws, size_t ws_size,
                              hipStream_t stream) {
    const float* x    = (const float*)d_in[0];
    const int*   ei   = (const int*)d_in[1];
    const int*   batch= (const int*)d_in[2];
    const float* W1   = (const float*)d_in[3];
    const float* b1   = (const float*)d_in[4];
    const float* W2   = (const float*)d_in[5];
    const float* b2   = (const float*)d_in[6];
    const float* Wmu  = (const float*)d_in[7];
    const float* bmu  = (const float*)d_in[8];
    const float* Wlv  = (const float*)d_in[9];
    const float* blv  = (const float*)d_in[10];
    float* out = (float*)d_out;

    const int N = in_sizes[0] / CH;           // 100000
    const int E = in_sizes[1] / 2;            // 1600000
    const int G = out_size / (2 * OUTCH);     // 256
    const int* src = ei;
    const int* dst = ei + E;

    float* h    = (float*)d_ws;               // [N, 128]
    float* agg  = h    + (size_t)N * CH;      // [N, 128]
    float* dinv = agg  + (size_t)N * CH;      // [N]
    float* sums = dinv + N;                   // [G, 128]
    float* cnts = sums + (size_t)G * CH;      // [G]

    dim3 blk(256);

    // symmetric normalization: deg (with self loop) -> rsqrt
    deg_init_kernel <<<(N + 255) / 256, blk, 0, stream>>>(dinv, N);
    deg_edges_kernel<<<(E + 255) / 256, blk, 0, stream>>>(dst, dinv, E);
    dinv_kernel     <<<(N + 255) / 256, blk, 0, stream>>>(dinv, N);

    // ---- layer 1 ----
    gemm_node_wmma  <<<(N + 15) / 16, blk, 0, stream>>>(x, W1, h, N);
    agg_init_kernel <<<(N * 32 + 255) / 256, blk, 0, stream>>>(h, dinv, agg, N);
    agg_edges_kernel<<<(E * 32 + 255) / 256, blk, 0, stream>>>(src, dst, h, dinv, agg, E);
    bias_relu_kernel<<<((long)N * CH + 255) / 256, blk, 0, stream>>>(agg, b1, N);

    // ---- layer 2 (agg -> h -> agg) ----
    gemm_node_wmma  <<<(N + 15) / 16, blk, 0, stream>>>(agg, W2, h, N);
    agg_init_kernel <<<(N * 32 + 255) / 256, blk, 0, stream>>>(h, dinv, agg, N);
    agg_edges_kernel<<<(E * 32 + 255) / 256, blk, 0, stream>>>(src, dst, h, dinv, agg, E);
    bias_relu_kernel<<<((long)N * CH + 255) / 256, blk, 0, stream>>>(agg, b2, N);

    // ---- global mean pool ----
    pool_zero_kernel <<<(G * CH + 255) / 256, blk, 0, stream>>>(sums, cnts, G);
    pool_accum_kernel<<<(N * 32 + 255) / 256, blk, 0, stream>>>(agg, batch, sums, cnts, N);
    pool_div_kernel  <<<(G * CH + 255) / 256, blk, 0, stream>>>(sums, cnts, G);

    // ---- heads ----
    head_gemm_wmma<<<(G + 15) / 16, blk, 0, stream>>>(sums, Wmu, bmu, Wlv, blv, out, G);
}